// differentiableSuperpixelEmbedding_56298431316559
// MI455X (gfx1250) — compile-verified
//
#include <hip/hip_runtime.h>

#define BB 4
#define CC 3
#define HH 224
#define WWID 224
#define HW (HH * WWID)        // 50176
#define KSEG 196
#define EE 768
#define CPAD 4                // pad C 3 -> 4 for V_WMMA_F32_16X16X4_F32
#define BLOCKS_PER_IMG 64
#define POOL_THREADS 256

#define MT ((BB * KSEG) / 16)   // 49 row tiles of the 784x768 output
#define NT (EE / 16)            // 48 col tiles
#define TILES (MT * NT)         // 2352
#define WAVES_PER_BLOCK 8

#define POOLED_ELTS (BB * KSEG * CPAD)   // 3136 f32
#define WPAD_ELTS   (CPAD * EE)          // 3072 f32
#define PREP_ELTS   (POOLED_ELTS + WPAD_ELTS)

typedef float v2f __attribute__((ext_vector_type(2)));
typedef float v8f __attribute__((ext_vector_type(8)));

// ---------------------------------------------------------------------------
// Kernel 1: per call, zero the padded pooled accumulator AND build the
// zero-padded weight matrix Wpad(4 x 768) so the GEMM loads unconditionally.
// ---------------------------------------------------------------------------
__global__ void prep_ws(const float* __restrict__ Wmat,  // (3, 768)
                        float* __restrict__ pooled,      // (B*K, CPAD)
                        float* __restrict__ Wpad) {      // (CPAD, EE)
    int i = blockIdx.x * blockDim.x + threadIdx.x;
    if (i < POOLED_ELTS) {
        pooled[i] = 0.0f;
    } else if (i < PREP_ELTS) {
        int j = i - POOLED_ELTS;        // j = k*EE + n
        Wpad[j] = (j < CC * EE) ? Wmat[j] : 0.0f;   // row 3 = zero pad
    }
}

// ---------------------------------------------------------------------------
// Kernel 2: segmented sum. LDS-resident K*CPAD accumulator per block,
// ds_add_f32 atomics per pixel, global f32 atomics only on flush.
// ---------------------------------------------------------------------------
__global__ __launch_bounds__(POOL_THREADS) void seg_pool(
    const float* __restrict__ img,   // (B, C, H, W)
    const int*   __restrict__ seg,   // (B, H, W)
    float*       __restrict__ pooled // (B, K, CPAD) sums (unscaled)
) {
    __shared__ float acc[KSEG * CPAD];   // 3136 B of LDS
    const int b   = blockIdx.x / BLOCKS_PER_IMG;
    const int blk = blockIdx.x % BLOCKS_PER_IMG;

    for (int i = threadIdx.x; i < KSEG * CPAD; i += POOL_THREADS) acc[i] = 0.0f;
    __syncthreads();

    const int per_blk = HW / BLOCKS_PER_IMG;   // 784 pixels
    const int p0 = blk * per_blk;
    const float* img_b = img + (size_t)b * CC * HW;
    const int*   seg_b = seg + (size_t)b * HW;

    for (int p = p0 + threadIdx.x; p < p0 + per_blk; p += POOL_THREADS) {
        const int   k  = seg_b[p];
        const float r  = img_b[0 * HW + p];
        const float g  = img_b[1 * HW + p];
        const float bl = img_b[2 * HW + p];
        atomicAdd(&acc[k * CPAD + 0], r);
        atomicAdd(&acc[k * CPAD + 1], g);
        atomicAdd(&acc[k * CPAD + 2], bl);
    }
    __syncthreads();

    for (int t = threadIdx.x; t < KSEG * CC; t += POOL_THREADS) {
        const int k = t / CC, c = t % CC;
        const float v = acc[k * CPAD + c];
        if (v != 0.0f) atomicAdd(&pooled[((size_t)b * KSEG + k) * CPAD + c], v);
    }
}

// ---------------------------------------------------------------------------
// Kernel 3: out(784x768) = (pooled * 1/HW)(784x4) @ Wpad(4x768) + bias,
// one V_WMMA_F32_16X16X4_F32 per wave (16x16 output tile, K=4 f32).
// Straight-line: no divergence anywhere, EXEC all-ones for the WMMA.
//
// Operand layouts (CDNA5 ISA 7.12.2, wave32):
//   A 16x4 f32 : lane(0-15)=M, v0=K0, v1=K1; lanes 16-31: v0=K2, v1=K3
//   B 4x16 f32 : lanes index N; lanes 0-15: v0=row K0, v1=row K1;
//                lanes 16-31: v0=row K2, v1=row K3
//   C/D 16x16  : VGPR i, lanes 0-15 -> M=i, lanes 16-31 -> M=i+8; N=lane&15
// ---------------------------------------------------------------------------
__global__ __launch_bounds__(WAVES_PER_BLOCK * 32) void wmma_gemm(
    const float* __restrict__ pooled,  // (784, CPAD) unscaled sums, col 3 == 0
    const float* __restrict__ Wpad,    // (CPAD, 768), row 3 == 0
    const float* __restrict__ bias,    // (768)
    float*       __restrict__ out      // (784, 768)
) {
    const int wave  = blockIdx.x * WAVES_PER_BLOCK + (threadIdx.x >> 5);
    const int lane  = threadIdx.x & 31;
    const int mtile = wave / NT;
    const int ntile = wave % NT;
    const int l15   = lane & 15;
    const int hi    = lane >> 4;          // 0: K=0,1 / M=0-7 ; 1: K=2,3 / M=8-15
    const float scale = 1.0f / (float)HW; // dense mean over all H*W pixels

    // A fragment (scale folded in; pooled col 3 is zero from kernels 1/2)
    const int m  = mtile * 16 + l15;
    const int kb = hi * 2;                // 0 or 2
    v2f a;
    a.x = pooled[(size_t)m * CPAD + kb + 0] * scale;
    a.y = pooled[(size_t)m * CPAD + kb + 1] * scale;

    // B fragment: unconditional loads from the zero-padded weights
    const int n = ntile * 16 + l15;
    v2f bm;
    bm.x = Wpad[(size_t)(kb + 0) * EE + n];
    bm.y = Wpad[(size_t)(kb + 1) * EE + n];

    // C = broadcast bias along M
    const float bv = bias[n];
    v8f c;
#pragma unroll
    for (int i = 0; i < 8; ++i) c[i] = bv;

    // D = A x B + C  (8 args: neg_a, A, neg_b, B, c_mod, C, reuse_a, reuse_b)
    v8f d = __builtin_amdgcn_wmma_f32_16x16x4_f32(
        false, a, false, bm, (short)0, c, false, false);

    const int mrow = mtile * 16 + hi * 8;
#pragma unroll
    for (int i = 0; i < 8; ++i) out[(size_t)(mrow + i) * EE + n] = d[i];
}

// ---------------------------------------------------------------------------
extern "C" void kernel_launch(void* const* d_in, const int* in_sizes, int n_in,
                              void* d_out, int out_size, void* d_ws, size_t ws_size,
                              hipStream_t stream) {
    const float* img  = (const float*)d_in[0];  // (4,3,224,224) f32
    const int*   seg  = (const int*)  d_in[1];  // (4,224,224) i32
    const float* Wmat = (const float*)d_in[2];  // (3,768) f32
    const float* bias = (const float*)d_in[3];  // (768) f32
    float* out    = (float*)d_out;              // (4,196,768) f32
    float* pooled = (float*)d_ws;               // (784, CPAD) f32 scratch
    float* Wpad   = pooled + POOLED_ELTS;       // (CPAD, 768) f32 scratch

    prep_ws<<<(PREP_ELTS + 255) / 256, 256, 0, stream>>>(Wmat, pooled, Wpad);
    seg_pool<<<BB * BLOCKS_PER_IMG, POOL_THREADS, 0, stream>>>(img, seg, pooled);
    wmma_gemm<<<TILES / WAVES_PER_BLOCK, WAVES_PER_BLOCK * 32, 0, stream>>>(
        pooled, Wpad, bias, out);
}